// Attention_4105988735881
// MI455X (gfx1250) — compile-verified
//
#include <hip/hip_runtime.h>
#include <math.h>

// ---------------------------------------------------------------------------
// MI455X (gfx1250) implementation. wave32, WMMA f32 16x16x4 matrix path.
// Shapes: B=8, Lq=Lk=160, D=H=512, C=8, S=64, R=8, OUT=512.
// All GEMM strides are compile-time template constants so addressing folds
// into immediate global_load offsets instead of v_mul_u64 chains.
// ---------------------------------------------------------------------------

typedef __attribute__((ext_vector_type(2))) float v2f;
typedef __attribute__((ext_vector_type(8))) float v8f;

static __device__ __forceinline__ v8f wmma4(v2f a, v2f b, v8f c) {
  // D = A(16x4) * B(4x16) + C(16x16), fp32 throughout.
  return __builtin_amdgcn_wmma_f32_16x16x4_f32(
      /*neg_a=*/false, a, /*neg_b=*/false, b,
      /*c_mod=*/(short)0, c, /*reuse_a=*/false, /*reuse_b=*/false);
}

static __device__ __forceinline__ v8f zero8() {
  v8f z = {0.f, 0.f, 0.f, 0.f, 0.f, 0.f, 0.f, 0.f};
  return z;
}

// ---------------------------------------------------------------------------
// Templated batched GEMM, one wave computes a 32x32 C tile (4 WMMA accums).
//   C[b,c][m,n] = sum_k A[b,c][m,k] * W[b,c][n,k] + bias[c][n]
// A element (m,k): A + b*SAB + c*KOFSC + m*LDAM + k            (k stride 1)
// W element (n,k): W + b*SWB + c*SWC + n*LDWN + k*LDWK
// C element (m,n): C + b*SCB + c*SCC + m*LDCM + n*LDCN
// Grid: (N/32, M/32, B*C2). Block: 32 (one wave). K % 8 == 0.
// ---------------------------------------------------------------------------
template <int LDAM, int KOFSC, long long SAB,
          long long SWB, long long SWC, int LDWN, int LDWK,
          bool HASB, int SBIASC,
          long long SCB, long long SCC, int LDCM, int LDCN,
          int K, int C2>
__global__ __launch_bounds__(32) void gemm_t(
    const float* __restrict__ A, const float* __restrict__ W,
    const float* __restrict__ bias, float* __restrict__ C)
{
  const int bz = blockIdx.z;
  const int b  = bz / C2;
  const int c  = bz % C2;
  const int n0 = blockIdx.x << 5;
  const int m0 = blockIdx.y << 5;
  const int lane = threadIdx.x & 31;
  const int ml = lane & 15;   // M row (A) / N col (B) within half-wave
  const int kh = lane >> 4;   // lanes 16-31 carry K=2,3 of each 4-group

  const float* A0 = A + (long long)b * SAB + (long long)c * KOFSC
                      + (long long)(m0 + ml) * LDAM;
  const float* A1 = A0 + 16LL * LDAM;
  const float* W0 = W + (long long)b * SWB + (long long)c * SWC
                      + (long long)(n0 + ml) * LDWN;
  const float* W1 = W0 + 16LL * LDWN;

  __builtin_prefetch(A0, 0, 1);
  __builtin_prefetch(W0, 0, 1);

  v8f acc00 = zero8(), acc01 = zero8(), acc10 = zero8(), acc11 = zero8();

#pragma unroll 2
  for (int k0 = 0; k0 < K; k0 += 8) {
    const int ka = k0 + (kh << 1);
    v2f a00 = *(const v2f*)(A0 + ka);
    v2f a01 = *(const v2f*)(A0 + ka + 4);
    v2f a10 = *(const v2f*)(A1 + ka);
    v2f a11 = *(const v2f*)(A1 + ka + 4);
    v2f b00, b01, b10, b11;
    b00.x = W0[(long long)(ka    ) * LDWK]; b00.y = W0[(long long)(ka + 1) * LDWK];
    b01.x = W0[(long long)(ka + 4) * LDWK]; b01.y = W0[(long long)(ka + 5) * LDWK];
    b10.x = W1[(long long)(ka    ) * LDWK]; b10.y = W1[(long long)(ka + 1) * LDWK];
    b11.x = W1[(long long)(ka + 4) * LDWK]; b11.y = W1[(long long)(ka + 5) * LDWK];
    acc00 = wmma4(a00, b00, acc00);
    acc00 = wmma4(a01, b01, acc00);
    acc01 = wmma4(a00, b10, acc01);
    acc01 = wmma4(a01, b11, acc01);
    acc10 = wmma4(a10, b00, acc10);
    acc10 = wmma4(a11, b01, acc10);
    acc11 = wmma4(a10, b10, acc11);
    acc11 = wmma4(a11, b11, acc11);
  }

  float bv0 = 0.f, bv1 = 0.f;
  if constexpr (HASB) {
    bv0 = bias[(long long)c * SBIASC + n0 + ml];
    bv1 = bias[(long long)c * SBIASC + n0 + 16 + ml];
  }
  float* Cb = C + (long long)b * SCB + (long long)c * SCC;
  const int mbase = m0 + (kh << 3);
#pragma unroll
  for (int e = 0; e < 8; ++e) {
    // C/D layout: VGPR e -> lanes 0-15: M=e, lanes 16-31: M=8+e; N=ml
    const long long r0 = (long long)(mbase + e) * LDCM;
    const long long r1 = (long long)(mbase + 16 + e) * LDCM;
    const long long cn0 = (long long)(n0 + ml) * LDCN;
    const long long cn1 = (long long)(n0 + 16 + ml) * LDCN;
    Cb[r0 + cn0] = acc00[e] + bv0;
    Cb[r0 + cn1] = acc01[e] + bv1;
    Cb[r1 + cn0] = acc10[e] + bv0;
    Cb[r1 + cn1] = acc11[e] + bv1;
  }
}

// ---------------------------------------------------------------------------
// Bilinear scores without materializing z (420MB avoided; all-L2 working set).
// y?t layout: [b][c][r*64+s][L], L (=q/k index, 160) contiguous -> every WMMA
// operand load is lane-contiguous global_load_b32.
// Per 16x16 (q,k) tile, per (c,s): z = Y1(16x8) * Y0(8x16) via two chained
// K=4 f32 WMMAs; reduce in-register to A_c = sum_s signsqrt(z)*w_bo and
// N_c = sum_s |z| (since signsqrt(z)^2 == |z|).
// score = sum_c A_c / max(sqrt(N_c), eps) + b_bo.
// Grid: (Lk/16, Lq/16, B). Block: 32.
// ---------------------------------------------------------------------------
__global__ __launch_bounds__(32) void block_scores_wmma(
    const float* __restrict__ y1t, const float* __restrict__ y0t,
    const float* __restrict__ w_bo, const float* __restrict__ b_bo,
    float* __restrict__ scores)
{
  const int b  = blockIdx.z;
  const int q0 = blockIdx.y << 4;
  const int k0 = blockIdx.x << 4;
  const int lane = threadIdx.x & 31;
  const int ml = lane & 15;
  const int kh = lane >> 4;
  const long long bOfs = (long long)b * (8LL * 512 * 160);
  const float* Y1 = y1t + bOfs;
  const float* Y0 = y0t + bOfs;

  float tot[8] = {0.f, 0.f, 0.f, 0.f, 0.f, 0.f, 0.f, 0.f};
  const int rbase = (kh << 1) * (64 * 160);  // lanes 0-15: r={0,1}; 16-31: r={2,3}
  constexpr int RS = 64 * 160;               // one r-plane (s-major) in floats

  for (int c = 0; c < 8; ++c) {
    const float* y1c = Y1 + (long long)c * (512 * 160) + rbase + q0 + ml;
    const float* y0c = Y0 + (long long)c * (512 * 160) + rbase + k0 + ml;
    __builtin_prefetch(y1c, 0, 1);
    __builtin_prefetch(y0c, 0, 1);
    float accA[8] = {0.f, 0.f, 0.f, 0.f, 0.f, 0.f, 0.f, 0.f};
    float accN[8] = {0.f, 0.f, 0.f, 0.f, 0.f, 0.f, 0.f, 0.f};

#pragma unroll 4
    for (int s = 0; s < 64; ++s) {
      const int so = s * 160;
      v2f a0, a1, bb0, bb1;
      a0.x  = y1c[so];                 // r = rb+0
      a0.y  = y1c[so + RS];            // r = rb+1
      a1.x  = y1c[so + 4 * RS];        // r = rb+4
      a1.y  = y1c[so + 5 * RS];        // r = rb+5
      bb0.x = y0c[so];
      bb0.y = y0c[so + RS];
      bb1.x = y0c[so + 4 * RS];
      bb1.y = y0c[so + 5 * RS];

      v8f z = zero8();
      z = wmma4(a0, bb0, z);           // rank contraction K=0..3
      z = wmma4(a1, bb1, z);           // rank contraction K=4..7

      const float w = w_bo[(c << 6) + s];
#pragma unroll
      for (int e = 0; e < 8; ++e) {
        const float zz = z[e];
        const float az = fabsf(zz);                       // |.| input modifier
        const float f  = copysignf(__builtin_amdgcn_sqrtf(az), zz);
        accA[e] += f * w;
        accN[e] += az;                                    // signsqrt(z)^2 == |z|
      }
    }
#pragma unroll
    for (int e = 0; e < 8; ++e)
      tot[e] += accA[e] / fmaxf(__builtin_amdgcn_sqrtf(accN[e]), 1e-12f);
  }

  const float bb = b_bo[0];
  float* outp = scores + ((long long)b * 160 + q0 + (kh << 3)) * 160 + k0 + ml;
#pragma unroll
  for (int e = 0; e < 8; ++e)
    outp[(long long)e * 160] = tot[e] + bb;
}

// ---------------------------------------------------------------------------
// Softmax over k (160) — one wave32 per (b,q) row, 5 elements/lane.
// ---------------------------------------------------------------------------
__global__ __launch_bounds__(32) void softmax160(float* __restrict__ p)
{
  float* row = p + (long long)blockIdx.x * 160;
  const int lane = threadIdx.x;
  float v[5];
  float mx = -3.402823466e38f;
#pragma unroll
  for (int i = 0; i < 5; ++i) { v[i] = row[lane + 32 * i]; mx = fmaxf(mx, v[i]); }
#pragma unroll
  for (int o = 16; o > 0; o >>= 1) mx = fmaxf(mx, __shfl_xor(mx, o, 32));
  float s = 0.f;
#pragma unroll
  for (int i = 0; i < 5; ++i) { v[i] = __expf(v[i] - mx); s += v[i]; }
#pragma unroll
  for (int o = 16; o > 0; o >>= 1) s += __shfl_xor(s, o, 32);
  const float inv = 1.f / s;
#pragma unroll
  for (int i = 0; i < 5; ++i) row[lane + 32 * i] = v[i] * inv;
}

// ---------------------------------------------------------------------------
// torch transpose(1,2).view(B,-1,H): mergeA[b, flat] = atted[b, flat%160, flat/160]
// ---------------------------------------------------------------------------
__global__ __launch_bounds__(256) void permute_merge(
    const float* __restrict__ atted, float* __restrict__ out)
{
  const int idx = blockIdx.x * 256 + threadIdx.x;
  if (idx >= 8 * 160 * 512) return;
  const int b   = idx / 81920;
  const int rem = idx - b * 81920;
  const int h   = rem / 160;
  const int l   = rem - h * 160;
  out[idx] = atted[b * 81920 + l * 512 + h];
}

// ---------------------------------------------------------------------------
extern "C" void kernel_launch(void* const* d_in, const int* in_sizes, int n_in,
                              void* d_out, int out_size, void* d_ws, size_t ws_size,
                              hipStream_t stream)
{
  (void)in_sizes; (void)n_in; (void)out_size; (void)ws_size;

  const float* v    = (const float*)d_in[0];
  const float* k    = (const float*)d_in[1];
  const float* q    = (const float*)d_in[2];
  const float* w_v  = (const float*)d_in[3];
  const float* b_v  = (const float*)d_in[4];
  const float* w_k  = (const float*)d_in[5];
  const float* b_k  = (const float*)d_in[6];
  const float* w_q  = (const float*)d_in[7];
  const float* b_q  = (const float*)d_in[8];
  const float* w0   = (const float*)d_in[9];
  const float* b0   = (const float*)d_in[10];
  const float* w1   = (const float*)d_in[11];
  const float* b1   = (const float*)d_in[12];
  const float* wm0  = (const float*)d_in[13];
  const float* bm0  = (const float*)d_in[14];
  const float* wm1  = (const float*)d_in[15];
  const float* bm1  = (const float*)d_in[16];
  const float* w_bo = (const float*)d_in[17];
  const float* b_bo = (const float*)d_in[18];
  const float* w_m  = (const float*)d_in[19];
  const float* b_m  = (const float*)d_in[20];

  // Workspace carve-up (floats). Total ~15.3M floats = ~61 MB (fits in L2).
  float* ws = (float*)d_ws;
  const long long BLH = 8LL * 160 * 512;      // 655360
  const long long YSZ = 8LL * 8 * 512 * 160;  // 5242880
  float* vv     = ws;
  float* kk     = vv + BLH;
  float* qq     = kk + BLH;
  float* x0     = qq + BLH;
  float* x1     = x0 + BLH;
  float* y0t    = x1 + BLH;        // [b][c][r*64+s][k]
  float* y1t    = y0t + YSZ;       // [b][c][r*64+s][q]
  float* sc     = y1t + YSZ;       // scores / att in-place, [b][q][k]
  float* atted  = sc + 8LL * 160 * 160;
  float* mergeA = atted + BLH;

  const dim3 wave(32);
  const dim3 gProj(16, 5, 8);      // N=512/32, M=160/32, B=8
  const dim3 gChunk(16, 5, 64);    // batched over b*c (C2=8)

  // Standard projection GEMM: M=160, N=512, K=512, shared weight [512,512].
  // <LDAM,KOFSC,SAB, SWB,SWC,LDWN,LDWK, HASB,SBIASC, SCB,SCC,LDCM,LDCN, K,C2>
  auto* proj = gemm_t<512, 0, 81920, 0, 0, 512, 1, true, 0,
                      81920, 0, 512, 1, 512, 1>;

  // 1-3) input projections
  proj<<<gProj, wave, 0, stream>>>(v, w_v, b_v, vv);
  proj<<<gProj, wave, 0, stream>>>(k, w_k, b_k, kk);
  proj<<<gProj, wave, 0, stream>>>(q, w_q, b_q, qq);

  // 4-5) block linears
  proj<<<gProj, wave, 0, stream>>>(kk, w0, b0, x0);
  proj<<<gProj, wave, 0, stream>>>(qq, w1, b1, x1);

  // 6-7) per-chunk merge GEMMs, output TRANSPOSED to [b][c][t=r*64+s][L]
  //      (M=160, N=512, K=64; A column window c*64; W/bias indexed by c)
  gemm_t<512, 64, 81920, 0, 32768, 64, 1, true, 512,
         655360, 81920, 1, 160, 64, 8>
      <<<gChunk, wave, 0, stream>>>(x0, wm0, bm0, y0t);
  gemm_t<512, 64, 81920, 0, 32768, 64, 1, true, 512,
         655360, 81920, 1, 160, 64, 8>
      <<<gChunk, wave, 0, stream>>>(x1, wm1, bm1, y1t);

  // 8) fused bilinear + signed-sqrt + chunk-L2-norm + w_bo projection
  block_scores_wmma<<<dim3(10, 10, 8), wave, 0, stream>>>(y1t, y0t, w_bo, b_bo, sc);

  // 9) softmax over k (in place)
  softmax160<<<dim3(8 * 160), wave, 0, stream>>>(sc);

  // 10) atted = att @ vv   (vv as B with ldwN=1, ldwK=512; K=160, no bias)
  gemm_t<160, 0, 25600, 81920, 0, 1, 512, false, 0,
         81920, 0, 512, 1, 160, 1>
      <<<gProj, wave, 0, stream>>>(sc, vv, nullptr, atted);

  // 11) torch transpose-view permutation
  permute_merge<<<dim3((8 * 160 * 512 + 255) / 256), dim3(256), 0, stream>>>(atted, mergeA);

  // 12) out = mergeA @ w_m^T + b_m  -> d_out [8,160,512]
  proj<<<gProj, wave, 0, stream>>>(mergeA, w_m, b_m, (float*)d_out);
}